// GATV2_87282325390039
// MI455X (gfx1250) — compile-verified
//
#include <hip/hip_runtime.h>
#include <hip/hip_bf16.h>

#define NFEAT 128
#define HC    128   // HEADS*NHID
#define HEADS 8
#define NHID  16
#define NCLASS 40
#define NEG_SLOPE 0.2f

typedef __attribute__((ext_vector_type(2))) float v2f;
typedef __attribute__((ext_vector_type(8))) float v8f;

__device__ __forceinline__ float lrelu(float x) { return x > 0.f ? x : NEG_SLOPE * x; }
__device__ __forceinline__ float elu(float x)   { return x > 0.f ? x : (__expf(x) - 1.f); }

__device__ __forceinline__ void atomicMaxF(float* addr, float v) {
    if (v >= 0.f) atomicMax((int*)addr, __float_as_int(v));
    else          atomicMin((unsigned int*)addr, __float_as_uint(v));
}

__device__ __forceinline__ void edge_sd(const int* __restrict__ ei, int E, int e, int& s, int& d) {
    if (e < E) { s = ei[e]; d = ei[E + e]; }
    else       { s = e - E; d = e - E; }
}

// ---------------- init ----------------
__global__ void k_init(float* s1, float* m1, float* out1, float* s2, float* m2,
                       float* outF, int N) {
    int gid = blockIdx.x * blockDim.x + threadIdx.x;
    if (gid < N * HC)  out1[gid] = 0.f;
    if (gid < N * NCLASS) outF[gid] = 0.f;
    if (gid < N * HEADS) { s1[gid] = 0.f; m1[gid] = -1e30f; }
    if (gid < N)       { s2[gid] = 0.f; m2[gid] = -1e30f; }
}

// ---------------- GEMM1: h1 = x @ W1  (Nx128 @ 128x128), fp32 WMMA ----------------
__global__ void k_gemm1(const float* __restrict__ A, const float* __restrict__ B,
                        float* __restrict__ C, int M) {
    const int wave = threadIdx.x >> 5;
    const int lane = threadIdx.x & 31;
    const int tile = blockIdx.x * 4 + wave;          // 16-row tile
    if (tile * 16 >= M) return;
    const int row   = tile * 16 + (lane & 15);
    const int khalf = (lane >> 4) * 2;               // 0 or 2
    const int col   = lane & 15;

    v8f acc[8];
#pragma unroll
    for (int t = 0; t < 8; ++t) acc[t] = (v8f){0.f,0.f,0.f,0.f,0.f,0.f,0.f,0.f};

    const float* arow = A + (size_t)row * NFEAT;
    for (int k = 0; k < NFEAT; k += 4) {
        v2f a;
        a.x = arow[k + khalf];
        a.y = arow[k + khalf + 1];
#pragma unroll
        for (int t = 0; t < 8; ++t) {
            v2f b;
            b.x = B[(k + khalf)     * HC + t * 16 + col];
            b.y = B[(k + khalf + 1) * HC + t * 16 + col];
            acc[t] = __builtin_amdgcn_wmma_f32_16x16x4_f32(
                false, a, false, b, (short)0, acc[t], false, false);
        }
    }
    const int rbase = tile * 16 + (lane >> 4) * 8;
#pragma unroll
    for (int t = 0; t < 8; ++t)
#pragma unroll
        for (int i = 0; i < 8; ++i)
            C[(size_t)(rbase + i) * HC + t * 16 + col] = acc[t][i];
}

// ---------------- attention logits layer 1: per (node, head) ----------------
__global__ void k_attn1(const float* __restrict__ h1, const float* __restrict__ att_src,
                        const float* __restrict__ att_dst,
                        float* __restrict__ a_src, float* __restrict__ a_dst, int N) {
    int gid = blockIdx.x * blockDim.x + threadIdx.x;
    if (gid >= N * HEADS) return;
    int n = gid >> 3, h = gid & 7;
    const float* hv = h1 + (size_t)n * HC + h * NHID;
    const float* as = att_src + h * NHID;
    const float* ad = att_dst + h * NHID;
    float ssum = 0.f, dsum = 0.f;
#pragma unroll
    for (int c = 0; c < NHID; ++c) { float v = hv[c]; ssum += v * as[c]; dsum += v * ad[c]; }
    a_src[gid] = ssum;
    a_dst[gid] = dsum;
}

// ---------------- edge passes layer 1 ----------------
__global__ void k_emax1(const int* __restrict__ ei, int E, int EE,
                        const float* __restrict__ a_src, const float* __restrict__ a_dst,
                        float* __restrict__ m1) {
    int e = blockIdx.x * blockDim.x + threadIdx.x;
    if (e >= EE) return;
    int s, d; edge_sd(ei, E, e, s, d);
#pragma unroll
    for (int h = 0; h < HEADS; ++h) {
        float el = lrelu(a_src[s * HEADS + h] + a_dst[d * HEADS + h]);
        atomicMaxF(&m1[d * HEADS + h], el);
    }
}

__global__ void k_esum1(const int* __restrict__ ei, int E, int EE,
                        const float* __restrict__ a_src, const float* __restrict__ a_dst,
                        const float* __restrict__ m1, float* __restrict__ ee1,
                        float* __restrict__ s1) {
    int e = blockIdx.x * blockDim.x + threadIdx.x;
    if (e >= EE) return;
    int s, d; edge_sd(ei, E, e, s, d);
#pragma unroll
    for (int h = 0; h < HEADS; ++h) {
        float el = lrelu(a_src[s * HEADS + h] + a_dst[d * HEADS + h]);
        float ev = __expf(el - m1[d * HEADS + h]);
        ee1[(size_t)e * HEADS + h] = ev;
        atomicAdd(&s1[d * HEADS + h], ev);
    }
}

// wave (32 lanes) per edge; lane covers 4 contiguous channels
__global__ void k_agg1(const int* __restrict__ ei, int E, int EE,
                       const float* __restrict__ h1, const float* __restrict__ ee1,
                       const float* __restrict__ s1, float* __restrict__ out1) {
    int gid = blockIdx.x * blockDim.x + threadIdx.x;
    int e = gid >> 5, lane = gid & 31;
    if (e >= EE) return;
    int s, d; edge_sd(ei, E, e, s, d);
    int h = lane >> 2;  // 16 channels per head, 4 lanes per head
    float alpha = ee1[(size_t)e * HEADS + h] / s1[d * HEADS + h];
    float4 hv = ((const float4*)(h1 + (size_t)s * HC))[lane];
    float* o = out1 + (size_t)d * HC + lane * 4;
    atomicAdd(o + 0, hv.x * alpha);
    atomicAdd(o + 1, hv.y * alpha);
    atomicAdd(o + 2, hv.z * alpha);
    atomicAdd(o + 3, hv.w * alpha);
}

__global__ void k_elu1(float* __restrict__ out1, const float* __restrict__ b1, int N) {
    int gid = blockIdx.x * blockDim.x + threadIdx.x;
    if (gid >= N * HC) return;
    out1[gid] = elu(out1[gid] + b1[gid & (HC - 1)]);
}

// ---------------- GEMM2: h2 = h1' @ W2  (Nx128 @ 128x40), fp32 WMMA, 3 col tiles ----------------
__global__ void k_gemm2(const float* __restrict__ A, const float* __restrict__ B,
                        float* __restrict__ C, int M) {
    const int wave = threadIdx.x >> 5;
    const int lane = threadIdx.x & 31;
    const int tile = blockIdx.x * 4 + wave;
    if (tile * 16 >= M) return;
    const int row   = tile * 16 + (lane & 15);
    const int khalf = (lane >> 4) * 2;
    const int col   = lane & 15;

    v8f acc[3];
#pragma unroll
    for (int t = 0; t < 3; ++t) acc[t] = (v8f){0.f,0.f,0.f,0.f,0.f,0.f,0.f,0.f};

    const float* arow = A + (size_t)row * HC;
    for (int k = 0; k < HC; k += 4) {
        v2f a;
        a.x = arow[k + khalf];
        a.y = arow[k + khalf + 1];
#pragma unroll
        for (int t = 0; t < 3; ++t) {
            int c = t * 16 + col;
            v2f b;
            b.x = (c < NCLASS) ? B[(k + khalf)     * NCLASS + c] : 0.f;
            b.y = (c < NCLASS) ? B[(k + khalf + 1) * NCLASS + c] : 0.f;
            acc[t] = __builtin_amdgcn_wmma_f32_16x16x4_f32(
                false, a, false, b, (short)0, acc[t], false, false);
        }
    }
    const int rbase = tile * 16 + (lane >> 4) * 8;
#pragma unroll
    for (int t = 0; t < 3; ++t) {
        int c = t * 16 + col;
        if (c < NCLASS) {
#pragma unroll
            for (int i = 0; i < 8; ++i)
                if (rbase + i < M)
                    C[(size_t)(rbase + i) * NCLASS + c] = acc[t][i];
        }
    }
}

__global__ void k_attn2(const float* __restrict__ h2, const float* __restrict__ att_src,
                        const float* __restrict__ att_dst,
                        float* __restrict__ a_src, float* __restrict__ a_dst, int N) {
    int n = blockIdx.x * blockDim.x + threadIdx.x;
    if (n >= N) return;
    const float* hv = h2 + (size_t)n * NCLASS;
    float ssum = 0.f, dsum = 0.f;
#pragma unroll
    for (int c = 0; c < NCLASS; ++c) { float v = hv[c]; ssum += v * att_src[c]; dsum += v * att_dst[c]; }
    a_src[n] = ssum;
    a_dst[n] = dsum;
}

__global__ void k_emax2(const int* __restrict__ ei, int E, int EE,
                        const float* __restrict__ a_src, const float* __restrict__ a_dst,
                        float* __restrict__ m2) {
    int e = blockIdx.x * blockDim.x + threadIdx.x;
    if (e >= EE) return;
    int s, d; edge_sd(ei, E, e, s, d);
    atomicMaxF(&m2[d], lrelu(a_src[s] + a_dst[d]));
}

__global__ void k_esum2(const int* __restrict__ ei, int E, int EE,
                        const float* __restrict__ a_src, const float* __restrict__ a_dst,
                        const float* __restrict__ m2, float* __restrict__ ee2,
                        float* __restrict__ s2) {
    int e = blockIdx.x * blockDim.x + threadIdx.x;
    if (e >= EE) return;
    int s, d; edge_sd(ei, E, e, s, d);
    float ev = __expf(lrelu(a_src[s] + a_dst[d]) - m2[d]);
    ee2[e] = ev;
    atomicAdd(&s2[d], ev);
}

__global__ void k_agg2(const int* __restrict__ ei, int E, int EE,
                       const float* __restrict__ h2, const float* __restrict__ ee2,
                       const float* __restrict__ s2, float* __restrict__ outF) {
    int gid = blockIdx.x * blockDim.x + threadIdx.x;
    int e = gid / NCLASS, c = gid - e * NCLASS;
    if (e >= EE) return;
    int s, d; edge_sd(ei, E, e, s, d);
    float alpha = ee2[e] / s2[d];
    atomicAdd(&outF[(size_t)d * NCLASS + c], h2[(size_t)s * NCLASS + c] * alpha);
}

// ---------------- final: elu + bias + log_softmax (in place on d_out) ----------------
__global__ void k_final(float* __restrict__ outF, const float* __restrict__ b2, int N) {
    int n = blockIdx.x * blockDim.x + threadIdx.x;
    if (n >= N) return;
    float v[NCLASS];
    float mx = -1e30f;
    float* o = outF + (size_t)n * NCLASS;
#pragma unroll
    for (int c = 0; c < NCLASS; ++c) { v[c] = elu(o[c] + b2[c]); mx = fmaxf(mx, v[c]); }
    float ss = 0.f;
#pragma unroll
    for (int c = 0; c < NCLASS; ++c) ss += __expf(v[c] - mx);
    float l = __logf(ss);
#pragma unroll
    for (int c = 0; c < NCLASS; ++c) o[c] = v[c] - mx - l;
}

extern "C" void kernel_launch(void* const* d_in, const int* in_sizes, int n_in,
                              void* d_out, int out_size, void* d_ws, size_t ws_size,
                              hipStream_t stream) {
    const float* x        = (const float*)d_in[0];
    const int*   ei       = (const int*)  d_in[2];
    const float* W1       = (const float*)d_in[3];
    const float* att_src1 = (const float*)d_in[4];
    const float* att_dst1 = (const float*)d_in[5];
    const float* b1       = (const float*)d_in[6];
    const float* W2       = (const float*)d_in[7];
    const float* att_src2 = (const float*)d_in[8];
    const float* att_dst2 = (const float*)d_in[9];
    const float* b2       = (const float*)d_in[10];

    const int N  = in_sizes[0] / NFEAT;
    const int E  = in_sizes[2] / 2;
    const int EE = E + N;
    float* outF = (float*)d_out;

    // workspace carve-up (floats)
    float* ws = (float*)d_ws;
    float* h1   = ws; ws += (size_t)N * HC;
    float* as1  = ws; ws += (size_t)N * HEADS;
    float* ad1  = ws; ws += (size_t)N * HEADS;
    float* m1   = ws; ws += (size_t)N * HEADS;
    float* s1   = ws; ws += (size_t)N * HEADS;
    float* ee1  = ws; ws += (size_t)EE * HEADS;
    float* out1 = ws; ws += (size_t)N * HC;
    float* h2   = ws; ws += (size_t)N * NCLASS;
    float* as2  = ws; ws += (size_t)N;
    float* ad2  = ws; ws += (size_t)N;
    float* m2   = ws; ws += (size_t)N;
    float* s2   = ws; ws += (size_t)N;
    float* ee2  = ws; ws += (size_t)EE;

    const int T = 256;
    const int tiles  = (N + 15) / 16;
    const int gblocks = (tiles + 3) / 4;

    k_init<<<(N * HC + T - 1) / T, T, 0, stream>>>(s1, m1, out1, s2, m2, outF, N);

    // layer 1
    k_gemm1<<<gblocks, 128, 0, stream>>>(x, W1, h1, N);
    k_attn1<<<(N * HEADS + T - 1) / T, T, 0, stream>>>(h1, att_src1, att_dst1, as1, ad1, N);
    k_emax1<<<(EE + T - 1) / T, T, 0, stream>>>(ei, E, EE, as1, ad1, m1);
    k_esum1<<<(EE + T - 1) / T, T, 0, stream>>>(ei, E, EE, as1, ad1, m1, ee1, s1);
    {
        long long thr = (long long)EE * 32;
        k_agg1<<<(unsigned)((thr + T - 1) / T), T, 0, stream>>>(ei, E, EE, h1, ee1, s1, out1);
    }
    k_elu1<<<(N * HC + T - 1) / T, T, 0, stream>>>(out1, b1, N);

    // layer 2
    k_gemm2<<<gblocks, 128, 0, stream>>>(out1, W2, h2, N);
    k_attn2<<<(N + T - 1) / T, T, 0, stream>>>(h2, att_src2, att_dst2, as2, ad2, N);
    k_emax2<<<(EE + T - 1) / T, T, 0, stream>>>(ei, E, EE, as2, ad2, m2);
    k_esum2<<<(EE + T - 1) / T, T, 0, stream>>>(ei, E, EE, as2, ad2, m2, ee2, s2);
    {
        long long thr = (long long)EE * NCLASS;
        k_agg2<<<(unsigned)((thr + T - 1) / T), T, 0, stream>>>(ei, E, EE, h2, ee2, s2, outF);
    }
    k_final<<<(N + T - 1) / T, T, 0, stream>>>(outF, b2, N);
}